// HOGLayerC_1211180778193
// MI455X (gfx1250) — compile-verified
//
#include <hip/hip_runtime.h>
#include <hip/hip_bf16.h>
#include <math.h>

#define NBINS 9
#define HIMG 224
#define WIMG 224
#define TILE 32
#define HALO 34          // TILE + 2 halo
#define CELLS 16         // 4x4 cells of 8x8 per tile
#define HPOOL 28
#define CN 27            // C*NBINS
#define KFEAT 108        // CN * 2 * 2

typedef __attribute__((ext_vector_type(2))) float v2f;
typedef __attribute__((ext_vector_type(8))) float v8f;

// ---------------------------------------------------------------------------
// Kernel 1: Sobel + orientation binning + gaussian weighting + 8x8 sum pool.
// One block = one 32x32 pixel tile of one (b,c) image plane.
// Halo tile staged into LDS with global_load_async_to_lds_b32 (ASYNCcnt path).
// Pooled histogram accumulated with ds_add_f32 LDS atomics.
// ---------------------------------------------------------------------------
__global__ __launch_bounds__(256) void hog_pool_kernel(
    const float* __restrict__ x, float* __restrict__ feat)
{
    __shared__ float tile[HALO * HALO];
    __shared__ float hist[CELLS * NBINS];
    __shared__ float gw[16];
    __shared__ float ginv;

    const int tid = threadIdx.x;
    const int t   = blockIdx.x;       // 0..48 tile index (7x7 tiles)
    const int bc  = blockIdx.y;       // 0..191 (b*3 + c)
    const int tx  = t % 7, ty = t / 7;
    const float* img = x + (size_t)bc * (HIMG * WIMG);

    // gaussian 1D weights (gkern row weights), std = 8, mean = 7.5
    if (tid < 16) {
        float n = ((float)tid - 7.5f) * 0.125f;
        gw[tid] = expf(-0.5f * n * n);
    }
    // zero the pooled histogram
    for (int idx = tid; idx < CELLS * NBINS; idx += 256)
        hist[idx] = 0.0f;

    // ---- async halo-tile load: global -> LDS, reflect padding at edges ----
    const unsigned ldsbase = (unsigned)(size_t)(void*)tile;
    for (int idx = tid; idx < HALO * HALO; idx += 256) {
        int r  = idx / HALO, c = idx % HALO;
        int gy = ty * TILE + r - 1;
        int gx = tx * TILE + c - 1;
        gy = (gy < 0) ? -gy : ((gy > HIMG - 1) ? 2 * (HIMG - 1) - gy : gy);
        gx = (gx < 0) ? -gx : ((gx > WIMG - 1) ? 2 * (WIMG - 1) - gx : gx);
        unsigned long long ga =
            (unsigned long long)(size_t)(img + gy * WIMG + gx);
        unsigned lo = ldsbase + (unsigned)idx * 4u;
        asm volatile("global_load_async_to_lds_b32 %0, %1, off"
                     :: "v"(lo), "v"(ga) : "memory");
    }
    asm volatile("s_wait_asynccnt 0" ::: "memory");
    __syncthreads();

    if (tid == 0) {
        float s = 0.0f;
        #pragma unroll
        for (int i = 0; i < 16; ++i) s += gw[i];
        ginv = 1.0f / (s * s);      // gkern normalization: g.sum() == (sum w)^2
    }
    __syncthreads();

    // ---- per-pixel gradient, bin, weight; LDS atomic scatter into cells ----
    const float gscale = ginv;
    #pragma unroll
    for (int i = 0; i < 4; ++i) {
        int p  = tid + 256 * i;          // 0..1023
        int py = p >> 5;                 // tile-local row
        int px = p & 31;                 // tile-local col
        const float* tc = tile + (py + 1) * HALO + (px + 1);
        float a00 = tc[-HALO - 1], a01 = tc[-HALO], a02 = tc[-HALO + 1];
        float a10 = tc[-1],                          a12 = tc[1];
        float a20 = tc[ HALO - 1], a21 = tc[ HALO],  a22 = tc[ HALO + 1];
        // cross-correlation with wx=[[1,0,-1],[2,0,-2],[1,0,-1]], wy=wx^T
        float gxv = (a00 + 2.0f * a10 + a20) - (a02 + 2.0f * a12 + a22);
        float gyv = (a00 + 2.0f * a01 + a02) - (a20 + 2.0f * a21 + a22);
        float nrm = sqrtf(gxv * gxv + gyv * gyv);
        // phase = atan2(gx, gy) / pi * NBINS ; bins = mod(floor(phase), 9)
        float ph = atan2f(gxv, gyv) * (9.0f / 3.14159265358979323846f);
        int bin = (int)floorf(ph);
        bin %= 9; if (bin < 0) bin += 9;
        float wv = nrm * gw[py & 15] * gw[px & 15] * gscale;
        int cell = (py >> 3) * 4 + (px >> 3);
        atomicAdd(&hist[cell * NBINS + bin], wv);   // ds_add_f32
    }
    __syncthreads();

    // ---- write pooled histograms: feat[b][cn=c*9+bin][hp][wp] ----
    for (int idx = tid; idx < CELLS * NBINS; idx += 256) {
        int cell = idx / NBINS, bin = idx % NBINS;
        int cy = cell >> 2, cx = cell & 3;
        int hp = ty * 4 + cy, wp = tx * 4 + cx;
        feat[(((size_t)bc * NBINS + bin) * HPOOL + hp) * HPOOL + wp] = hist[idx];
    }
}

// ---------------------------------------------------------------------------
// Kernel 2: per-patch standardization + mean.
// One wave32 per group of 16 patches of one batch image.
// Sum and sum-of-squares over the 108 features computed with
// V_WMMA_F32_16X16X4_F32 against an all-ones B matrix (row-sum reduction).
// ---------------------------------------------------------------------------
__global__ __launch_bounds__(32) void hog_norm_kernel(
    const float* __restrict__ feat, float* __restrict__ out)
{
    __shared__ float fbuf[16][KFEAT];   // 16 patches x 108 features
    __shared__ float sums[16];
    __shared__ float sqs[16];

    const int lane = threadIdx.x;       // 0..31
    const int g    = blockIdx.x;        // 0..12 patch group
    const int b    = blockIdx.y;        // 0..63

    // ---- gather the 16x108 f-block into LDS (zeros for padded patches) ----
    for (int e = lane; e < 16 * KFEAT; e += 32) {
        int m = e / KFEAT, k = e % KFEAT;
        int p = g * 16 + m;
        float v = 0.0f;
        if (p < 196) {
            int cn = k >> 2, ui = (k >> 1) & 1, uj = k & 1;
            int i2 = p / 14, j2 = p % 14;
            int hp = i2 * 2 + ui, wp = j2 * 2 + uj;
            v = feat[(((size_t)b * CN + cn) * HPOOL + hp) * HPOOL + wp];
        }
        fbuf[m][k] = v;
    }
    __syncthreads();

    // ---- WMMA reduction: D = A x ones + D  => every column = row sum ----
    // A-matrix f32 16x4 layout: lane l holds row m = l%16;
    //   VGPR0 = K = 2*(l/16), VGPR1 = K = 2*(l/16)+1
    const int m  = lane & 15;
    const int kh = lane >> 4;
    v8f dsum = {};
    v8f dsq  = {};
    v2f ones; ones[0] = 1.0f; ones[1] = 1.0f;
    #pragma unroll
    for (int kk = 0; kk < 27; ++kk) {
        int k0 = kk * 4 + kh * 2;
        v2f a;  a[0] = fbuf[m][k0];       a[1] = fbuf[m][k0 + 1];
        v2f a2; a2[0] = a[0] * a[0];      a2[1] = a[1] * a[1];
        dsum = __builtin_amdgcn_wmma_f32_16x16x4_f32(
            false, a,  false, ones, (short)0, dsum, false, false);
        dsq  = __builtin_amdgcn_wmma_f32_16x16x4_f32(
            false, a2, false, ones, (short)0, dsq,  false, false);
    }

    // D layout: VGPR v, lanes 0-15 -> M=v ; lanes 16-31 -> M=8+v.
    // Column 0 replica lives in lane 0 / lane 16 — they publish the 16 sums.
    if ((lane & 15) == 0) {
        int half = lane >> 4;
        #pragma unroll
        for (int v = 0; v < 8; ++v) {
            sums[half * 8 + v] = dsum[v];
            sqs [half * 8 + v] = dsq[v];
        }
    }
    __syncthreads();

    // ---- faithful normalize + mean (unbiased var, eps=1e-6) ----
    if (lane < 16) {
        int p = g * 16 + lane;
        if (p < 196) {
            float s   = sums[lane], ss = sqs[lane];
            float mu  = s * (1.0f / 108.0f);
            float var = (ss - s * mu) * (1.0f / 107.0f);
            float rstd = rsqrtf(var + 1e-6f);
            float acc = 0.0f;
            for (int k = 0; k < KFEAT; ++k)
                acc += (fbuf[lane][k] - mu);
            out[(size_t)b * 196 + p] = acc * rstd * (1.0f / 108.0f);
        }
    }
}

// ---------------------------------------------------------------------------
extern "C" void kernel_launch(void* const* d_in, const int* in_sizes, int n_in,
                              void* d_out, int out_size, void* d_ws, size_t ws_size,
                              hipStream_t stream) {
    (void)in_sizes; (void)n_in; (void)out_size; (void)ws_size;
    const float* x   = (const float*)d_in[0];
    float*       out = (float*)d_out;
    float*       ftw = (float*)d_ws;   // needs 64*27*28*28*4 B ~= 5.2 MB

    // Stage 1: gradients + binning + 8x8 pooling -> feat[64,27,28,28]
    hog_pool_kernel<<<dim3(49, 192), 256, 0, stream>>>(x, ftw);
    // Stage 2: per-patch standardization via WMMA reductions -> out[64,196]
    hog_norm_kernel<<<dim3(13, 64), 32, 0, stream>>>(ftw, out);
}